// GATModel_80418967651001
// MI455X (gfx1250) — compile-verified
//
#include <hip/hip_runtime.h>
#include <hip/hip_bf16.h>

typedef __attribute__((ext_vector_type(2))) float v2f;
typedef __attribute__((ext_vector_type(8))) float v8f;

#define Bq 64
#define Nn 256
#define Ee 2048
#define Dd 128
#define Hh 4
#define MAXE (Ee + 1)
#define NEG_SLOPE 0.2f

// ---------------------------------------------------------------------------
// Kernel 1: fold attention vectors through W once per conv:
//   c[h][k] = sum_d a[h][d] * W[h*128+d][k]     (c_src and c_dst, both convs)
// ---------------------------------------------------------------------------
__global__ void gat_prep(const float* __restrict__ W1, const float* __restrict__ as1,
                         const float* __restrict__ ad1,
                         const float* __restrict__ W2, const float* __restrict__ as2,
                         const float* __restrict__ ad2,
                         float* __restrict__ wsC) {
  const int c = blockIdx.x >> 1, t = blockIdx.x & 1;
  const float* W = c ? W2 : W1;
  const float* a = c ? (t ? ad2 : as2) : (t ? ad1 : as1);
  float* out = wsC + c * 1024 + t * 512;
  const int k = threadIdx.x;  // 0..127
  for (int h = 0; h < Hh; ++h) {
    float s = 0.f;
    const float* Wh = W + (h * Dd) * Dd + k;
    const float* ah = a + h * Dd;
    for (int d = 0; d < Dd; ++d) s += ah[d] * Wh[d * Dd];
    out[h * Dd + k] = s;
  }
}

// ---------------------------------------------------------------------------
// Kernel 2: one workgroup per (graph b, conv c). Computes node-0 GAT output.
// ---------------------------------------------------------------------------
__global__ __launch_bounds__(256) void gat_center(
    const int* __restrict__ nn1, const int* __restrict__ nn2,
    const int* __restrict__ adj1, const int* __restrict__ adj2,
    const float* __restrict__ emb,
    const float* __restrict__ W1, const float* __restrict__ b1,
    const float* __restrict__ W2, const float* __restrict__ b2,
    const float* __restrict__ wsC, float* __restrict__ feats) {
  __shared__ float x0s[Dd];          // center node embedding
  __shared__ float cSs[512];         // staged c_src
  __shared__ float adst0s[Hh];
  __shared__ int   srcs[MAXE];       // sources of edges into node 0 (+self loop)
  __shared__ float evals[MAXE * Hh]; // e-values, then alpha in place
  __shared__ float zpart[8 * 512];   // per-wave partials; REUSED as padded B [16x128]
  __shared__ float outf[512];        // W_h @ z[h]  (also scratch for adst0)
  __shared__ int   flags[257];
  __shared__ int   cntTot;

  const int tid  = threadIdx.x;
  const int lane = tid & 31;
  const int w    = tid >> 5;                 // wave id, 8 wave32s
  const int b    = blockIdx.x >> 1, c = blockIdx.x & 1;

  const int* nodes   = (c ? nn2 : nn1) + b * Nn;
  const int* adj     = (c ? adj2 : adj1) + b * 2 * Ee;
  const float* W     = c ? W2 : W1;
  const float* bias  = c ? b2 : b1;
  const float* csrc  = wsC + c * 1024;
  const float* cdst  = wsC + c * 1024 + 512;

  // --- stage center embedding and folded c_src ---
  if (tid < Dd) x0s[tid] = emb[(size_t)nodes[0] * Dd + tid];
  cSs[tid]       = csrc[tid];
  cSs[tid + 256] = csrc[tid + 256];
  __syncthreads();

  // --- adst0[h] = x0 . c_dst[h] ---
  outf[tid]       = x0s[tid & 127] * cdst[tid];
  outf[tid + 256] = x0s[tid & 127] * cdst[tid + 256];
  __syncthreads();
  if (tid < Hh) {
    float s = 0.f;
    for (int k = 0; k < Dd; ++k) s += outf[tid * Dd + k];
    adst0s[tid] = s;
  }

  // --- deterministic compaction of edges with dst==0 (prefix scan) ---
  int localsrc[8];
  int nloc = 0;
  {
    const int e0 = tid * 8;
    for (int j = 0; j < 8; ++j) {
      const int e = e0 + j;
      if (adj[Ee + e] == 0) localsrc[nloc++] = adj[e];
    }
  }
  flags[tid + 1] = nloc;
  if (tid == 0) flags[0] = 0;
  __syncthreads();
  for (int off = 1; off < 256; off <<= 1) {
    const int v   = flags[tid + 1];
    const int add = (tid >= off) ? flags[tid + 1 - off] : 0;
    __syncthreads();
    flags[tid + 1] = v + add;
    __syncthreads();
  }
  {
    const int start = flags[tid];
    for (int j = 0; j < nloc; ++j) srcs[start + j] = localsrc[j];
  }
  __syncthreads();
  if (tid == 0) { const int tot = flags[256]; srcs[tot] = 0; cntTot = tot + 1; }
  __syncthreads();
  const int ne = cntTot;

  // --- phase 1: e[i,h] = leakyrelu(x[src_i].c_src[h] + adst0[h]) ; one wave per edge ---
  for (int i = w; i < ne; i += 8) {
    const float* xs = emb + (size_t)srcs[i] * Dd;
    const float x0v = xs[lane], x1v = xs[lane + 32], x2v = xs[lane + 64], x3v = xs[lane + 96];
    float p[Hh];
    for (int h = 0; h < Hh; ++h)
      p[h] = x0v * cSs[h * Dd + lane]      + x1v * cSs[h * Dd + lane + 32]
           + x2v * cSs[h * Dd + lane + 64] + x3v * cSs[h * Dd + lane + 96];
    for (int off = 16; off > 0; off >>= 1)
      for (int h = 0; h < Hh; ++h) p[h] += __shfl_down(p[h], off, 32);
    if (lane == 0)
      for (int h = 0; h < Hh; ++h) {
        const float t = p[h] + adst0s[h];
        evals[i * Hh + h] = t > 0.f ? t : NEG_SLOPE * t;
      }
  }
  __syncthreads();

  // --- softmax over incoming edges, per head (serial, deg ~ 9) ---
  if (tid < Hh) {
    const int h = tid;
    float m = -1e30f;
    for (int i = 0; i < ne; ++i) m = fmaxf(m, evals[i * Hh + h]);
    float s = 0.f;
    for (int i = 0; i < ne; ++i) s += expf(evals[i * Hh + h] - m);
    const float inv = 1.f / s;
    for (int i = 0; i < ne; ++i) evals[i * Hh + h] = expf(evals[i * Hh + h] - m) * inv;
  }
  __syncthreads();

  // --- phase 3: z[h] = sum_e alpha[e,h] * x[src_e]  (register partials, fixed order) ---
  float zp[16];
  for (int q = 0; q < 16; ++q) zp[q] = 0.f;
  for (int i = w; i < ne; i += 8) {
    const float* xs = emb + (size_t)srcs[i] * Dd;
    const float xv0 = xs[lane], xv1 = xs[lane + 32], xv2 = xs[lane + 64], xv3 = xs[lane + 96];
    for (int h = 0; h < Hh; ++h) {
      const float a = evals[i * Hh + h];
      zp[h * 4 + 0] += a * xv0; zp[h * 4 + 1] += a * xv1;
      zp[h * 4 + 2] += a * xv2; zp[h * 4 + 3] += a * xv3;
    }
  }
  for (int h = 0; h < Hh; ++h)
    for (int j = 0; j < 4; ++j)
      zpart[w * 512 + h * Dd + lane + 32 * j] = zp[h * 4 + j];
  __syncthreads();

  // --- combine partials into registers, then REUSE zpart[0..2047] as the
  //     fully padded B matrix Zpad[16][128] (rows 0..3 = z[h], rows 4..15 = 0)
  //     so the WMMA B read is unconditional (no EXEC-divergent ds loads). ---
  float s0 = 0.f, s1 = 0.f;
  for (int wv = 0; wv < 8; ++wv) {
    s0 += zpart[wv * 512 + tid];
    s1 += zpart[wv * 512 + tid + 256];
  }
  __syncthreads();                 // all reads of partials done before overwrite
  float* zpad = zpart;             // [16][128], 2048 floats
  zpad[tid]       = s0;            // rows 0..1  (h*128+k, tid in [0,256))
  zpad[tid + 256] = s1;            // rows 2..3
  for (int j = 0; j < 6; ++j)      // rows 4..15 = zero padding
    zpad[512 + j * 256 + tid] = 0.f;
  __syncthreads();

  // --- phase 4: out0 = W @ Zpad via V_WMMA_F32_16X16X4_F32 (fp32 matrix pipe) ---
  // A = W [512x128], B = Zpad [128x16]. 32 M-tiles over 8 waves; 32 K=4 steps.
  {
    const int khalf = lane >> 4;         // A/B VGPR K split: {0,2}/{1,3}
    const int n16   = lane & 15;
    for (int mt = w * 4; mt < w * 4 + 4; ++mt) {
      const float* Wr = W + (mt * 16 + n16) * Dd;   // lane's A row (M = n16)
      v8f acc = {};
      for (int kt = 0; kt < 32; ++kt) {
        const int kk = kt * 4 + khalf * 2;
        const v2f av = *(const v2f*)(Wr + kk);              // global_load_b64
        const v2f bv = *(const v2f*)(zpad + n16 * Dd + kk); // ds_load_b64, no divergence
        acc = __builtin_amdgcn_wmma_f32_16x16x4_f32(
            false, av, false, bv, (short)0, acc, false, false);
      }
      // rows of tile mt all belong to head h = mt>>3; keep D column N == h only
      const int h = mt >> 3;
      if (n16 == h) {
        const int base = mt * 16 + (lane < 16 ? 0 : 8);   // D: VGPR r = rows r / r+8
        for (int r = 0; r < 8; ++r) outf[base + r] = acc[r];
      }
    }
  }
  __syncthreads();

  // --- mean over heads + bias -> feats[b, c*128 + d] ---
  if (tid < Dd) {
    const float f = 0.25f * (outf[tid] + outf[Dd + tid] + outf[2 * Dd + tid] + outf[3 * Dd + tid])
                  + bias[tid];
    feats[b * 256 + c * Dd + tid] = f;
  }
}

// ---------------------------------------------------------------------------
// Kernel 3: y[b] = feats[b,:] . mlp_w + mlp_b
// ---------------------------------------------------------------------------
__global__ void gat_final(const float* __restrict__ feats, const float* __restrict__ mw,
                          const float* __restrict__ mb, float* __restrict__ out) {
  __shared__ float red[256];
  const int b = blockIdx.x, tid = threadIdx.x;
  red[tid] = feats[b * 256 + tid] * mw[tid];
  __syncthreads();
  for (int off = 128; off > 0; off >>= 1) {
    if (tid < off) red[tid] += red[tid + off];
    __syncthreads();
  }
  if (tid == 0) out[b] = red[0] + mb[0];
}

extern "C" void kernel_launch(void* const* d_in, const int* in_sizes, int n_in,
                              void* d_out, int out_size, void* d_ws, size_t ws_size,
                              hipStream_t stream) {
  const int*   nn1  = (const int*)d_in[0];
  const int*   nn2  = (const int*)d_in[1];
  const int*   adj1 = (const int*)d_in[2];
  const int*   adj2 = (const int*)d_in[3];
  const float* emb  = (const float*)d_in[4];
  const float* W1   = (const float*)d_in[5];
  const float* as1  = (const float*)d_in[6];
  const float* ad1  = (const float*)d_in[7];
  const float* b1   = (const float*)d_in[8];
  const float* W2   = (const float*)d_in[9];
  const float* as2  = (const float*)d_in[10];
  const float* ad2  = (const float*)d_in[11];
  const float* b2   = (const float*)d_in[12];
  const float* mw   = (const float*)d_in[13];
  const float* mb   = (const float*)d_in[14];

  float* wsC   = (float*)d_ws;    // 2048 floats: folded attention (2 convs x src/dst x [4,128])
  float* feats = wsC + 2048;      // 64 x 256 floats

  gat_prep  <<<4,      128, 0, stream>>>(W1, as1, ad1, W2, as2, ad2, wsC);
  gat_center<<<Bq * 2, 256, 0, stream>>>(nn1, nn2, adj1, adj2, emb, W1, b1, W2, b2, wsC, feats);
  gat_final <<<Bq,     256, 0, stream>>>(feats, mw, mb, (float*)d_out);
}